// GAT_45827301048579
// MI455X (gfx1250) — compile-verified
//
#include <hip/hip_runtime.h>
#include <hip/hip_bf16.h>
#include <math.h>

// ---------------- problem constants (match reference) ----------------
#define IN_F   64
#define OUT_F  128
#define EDGEF  16
#define HIDF   32
#define NNODE  20000
#define NEDGE  50000
#define NGRAPH 512
#define KDIM   64
// GEMM output columns: 16*128 (edge_W part) + 128 (edge_b part)
#define NCOL   2176
#define NTILE_COL (NCOL / 16)     // 136
#define NTILE_ROW (NNODE / 16)    // 1250 (20000 divisible by 16)

typedef __attribute__((ext_vector_type(2))) float v2f;
typedef __attribute__((ext_vector_type(8))) float v8f;

// ---------------- kernel 0: repack [edge_W | edge_b] into B [64 x 2176] ----
// Bmat[i, f*128+o] = edge_W[f, i*128+o];  Bmat[i, 2048+o] = edge_b[i*128+o]
__global__ void prep_b_kernel(const float* __restrict__ edge_W,
                              const float* __restrict__ edge_b,
                              float* __restrict__ Bmat) {
  int idx = blockIdx.x * blockDim.x + threadIdx.x;
  if (idx >= IN_F * NCOL) return;
  int i = idx / NCOL;
  int c = idx - i * NCOL;
  float v;
  if (c < EDGEF * OUT_F) {
    int f = c >> 7;          // c / 128
    int o = c & (OUT_F - 1); // c % 128
    v = edge_W[f * (IN_F * OUT_F) + i * OUT_F + o];
  } else {
    int o = c - EDGEF * OUT_F;
    v = edge_b[i * OUT_F + o];
  }
  Bmat[idx] = v;
}

// ---------------- kernel 1: P = n @ Bmat  via V_WMMA_F32_16X16X4_F32 -------
// One wave per 16x16 tile, 8 waves (8 tiles) per 256-thread block.
__global__ __launch_bounds__(256) void gemm_p_kernel(const float* __restrict__ A,   // [NNODE x 64]
                                                     const float* __restrict__ B,   // [64 x NCOL]
                                                     float* __restrict__ P) {       // [NNODE x NCOL]
  const int lane = threadIdx.x & 31;
  const int wave = threadIdx.x >> 5;
  const int tile = blockIdx.x * 8 + wave;
  const int tm = tile / NTILE_COL;
  const int tn = tile - tm * NTILE_COL;
  const int m0 = tm * 16;
  const int n0 = tn * 16;
  const int r  = lane & 15;   // row (A) / col (B,C)
  const int hi = lane >> 4;   // selects K pair 0..1 / 2..3 and C row halves

  const float* Arow = A + (size_t)(m0 + r) * IN_F;

  v8f c = {};
#pragma unroll
  for (int k0 = 0; k0 < KDIM; k0 += 4) {
    const int ka = k0 + 2 * hi;
    v2f a, b;
    // A 16x4 fp32: VGPR0 = K even, VGPR1 = K odd; lanes16-31 hold K+2
    a.x = Arow[ka];
    a.y = Arow[ka + 1];
    // B 4x16 fp32: row striped across lanes within a VGPR
    b.x = B[(size_t)ka * NCOL + n0 + r];
    b.y = B[(size_t)(ka + 1) * NCOL + n0 + r];
    // (neg_a, A, neg_b, B, c_mod, C, reuse_a, reuse_b)
    c = __builtin_amdgcn_wmma_f32_16x16x4_f32(false, a, false, b, (short)0, c,
                                              false, false);
  }
  // C/D 16x16 f32: VGPR rr -> row rr (lanes 0-15) / row rr+8 (lanes 16-31)
#pragma unroll
  for (int rr = 0; rr < 8; ++rr) {
    int row = m0 + rr + hi * 8;
    P[(size_t)row * NCOL + n0 + r] = c[rr];
  }
}

// ---------------- kernel 2: init h = conv_b broadcast ----------------------
__global__ void init_h_kernel(const float* __restrict__ conv_b, float* __restrict__ h) {
  int idx = blockIdx.x * blockDim.x + threadIdx.x;
  if (idx < NNODE * OUT_F) h[idx] = conv_b[idx & (OUT_F - 1)];
}

// init num=0, denom=0, gmax=-inf
__global__ void init_pool_kernel(float* __restrict__ num, float* __restrict__ denom,
                                 float* __restrict__ gmax) {
  int idx = blockIdx.x * blockDim.x + threadIdx.x;
  if (idx < NGRAPH * OUT_F) num[idx] = 0.f;
  if (idx < NGRAPH) { denom[idx] = 0.f; gmax[idx] = -INFINITY; }
}

// ---------------- kernel 3: per-edge combine + scatter to h ----------------
// m[e,o] = P[src, 2048+o] + sum_f e[e,f] * P[src, f*128+o];  h[dst,o] += m
__global__ __launch_bounds__(128) void edge_kernel(const float* __restrict__ ef,
                                                   const int* __restrict__ src,
                                                   const int* __restrict__ dst,
                                                   const float* __restrict__ P,
                                                   float* __restrict__ h) {
  __shared__ float se[EDGEF];
  const int e = blockIdx.x;
  const int o = threadIdx.x;
  if (o < EDGEF) se[o] = ef[e * EDGEF + o];
  __syncthreads();
  const int s = src[e];
  const int d = dst[e];
  const float* Pr = P + (size_t)s * NCOL;
  float m = Pr[EDGEF * OUT_F + o];   // n_src @ edge_b term
#pragma unroll
  for (int f = 0; f < EDGEF; ++f) m += se[f] * Pr[f * OUT_F + o];
  atomicAdd(&h[(size_t)d * OUT_F + o], m);
}

// ---------------- kernel 4: gate scores + segment max ----------------------
__global__ void gate_kernel(const float* __restrict__ h,
                            const float* __restrict__ gate_W,
                            const float* __restrict__ gate_b,
                            const int* __restrict__ n2g,
                            float* __restrict__ gate, float* __restrict__ gmax) {
  int v = blockIdx.x * blockDim.x + threadIdx.x;
  if (v >= NNODE) return;
  const float* hr = h + (size_t)v * OUT_F;
  float acc = gate_b[0];
#pragma unroll 8
  for (int o = 0; o < OUT_F; ++o) acc += hr[o] * gate_W[o];
  gate[v] = acc;
  // float atomic-max via monotone int/uint bit trick
  float* addr = gmax + n2g[v];
  if (acc >= 0.f) atomicMax((int*)addr, __float_as_int(acc));
  else            atomicMin((unsigned int*)addr, __float_as_uint(acc));
}

// guard empty graphs: non-finite max -> 0 (matches reference)
__global__ void gmax_fix_kernel(float* __restrict__ gmax) {
  int g = blockIdx.x * blockDim.x + threadIdx.x;
  if (g < NGRAPH && !isfinite(gmax[g])) gmax[g] = 0.f;
}

// ---------------- kernel 5: attention accumulate (one pass) ----------------
// a = exp(gate - gmax[g]); denom[g]+=a; num[g,:] += a*h[v,:]
__global__ __launch_bounds__(128) void attn_kernel(const float* __restrict__ h,
                                                   const float* __restrict__ gate,
                                                   const float* __restrict__ gmax,
                                                   const int* __restrict__ n2g,
                                                   float* __restrict__ denom,
                                                   float* __restrict__ num) {
  const int v = blockIdx.x;
  const int o = threadIdx.x;
  const int g = n2g[v];
  const float a = expf(gate[v] - gmax[g]);
  if (o == 0) atomicAdd(&denom[g], a);
  atomicAdd(&num[(size_t)g * OUT_F + o], a * h[(size_t)v * OUT_F + o]);
}

// ---------------- kernel 6: pooled -> ELU -> MLP (one wave per graph) ------
__global__ __launch_bounds__(32) void final_kernel(const float* __restrict__ num,
                                                   const float* __restrict__ denom,
                                                   const float* __restrict__ W1,
                                                   const float* __restrict__ b1,
                                                   const float* __restrict__ W2,
                                                   const float* __restrict__ b2,
                                                   float* __restrict__ out) {
  const int g = blockIdx.x;
  const int j = threadIdx.x;        // hidden unit, 0..31 (one wave32)
  const float dnm = denom[g];
  const float inv = dnm > 0.f ? 1.f / dnm : 0.f;   // empty graph -> pooled 0
  float acc = b1[j];
  for (int o = 0; o < OUT_F; ++o) {
    float p = num[(size_t)g * OUT_F + o] * inv;    // broadcast load
    float z = p > 0.f ? p : expm1f(p);             // ELU
    acc += z * W1[o * HIDF + j];
  }
  float part = fmaxf(acc, 0.f) * W2[j];            // ReLU then W2
#pragma unroll
  for (int off = 16; off > 0; off >>= 1) part += __shfl_down(part, off, 32);
  if (j == 0) out[g] = part + b2[0];
}

// ---------------- host-side launch -----------------------------------------
extern "C" void kernel_launch(void* const* d_in, const int* in_sizes, int n_in,
                              void* d_out, int out_size, void* d_ws, size_t ws_size,
                              hipStream_t stream) {
  const float* n_feat  = (const float*)d_in[0];   // [N,64]
  const float* e_feat  = (const float*)d_in[1];   // [E,16]
  const int*   src     = (const int*)  d_in[2];
  const int*   dst     = (const int*)  d_in[3];
  const int*   n2g     = (const int*)  d_in[4];
  const float* edge_W  = (const float*)d_in[5];   // [16,8192]
  const float* edge_b  = (const float*)d_in[6];   // [8192]
  const float* conv_b  = (const float*)d_in[7];   // [128]
  const float* gate_W  = (const float*)d_in[8];   // [128]
  const float* gate_b  = (const float*)d_in[9];   // [1]
  const float* W1      = (const float*)d_in[10];  // [128,32]
  const float* b1      = (const float*)d_in[11];  // [32]
  const float* W2      = (const float*)d_in[12];  // [32]
  const float* b2      = (const float*)d_in[13];  // [1]
  float* out = (float*)d_out;                     // [512]

  // workspace carve-up (~185 MB)
  float* Bmat  = (float*)d_ws;                            // 64*2176
  float* P     = Bmat  + (size_t)IN_F * NCOL;             // 20000*2176
  float* h     = P     + (size_t)NNODE * NCOL;            // 20000*128
  float* gate  = h     + (size_t)NNODE * OUT_F;           // 20000
  float* gmax  = gate  + NNODE;                           // 512
  float* denom = gmax  + NGRAPH;                          // 512
  float* num   = denom + NGRAPH;                          // 512*128

  // 0) repack weights
  {
    int total = IN_F * NCOL;
    prep_b_kernel<<<(total + 255) / 256, 256, 0, stream>>>(edge_W, edge_b, Bmat);
  }
  // inits
  {
    int total = NNODE * OUT_F;
    init_h_kernel<<<(total + 255) / 256, 256, 0, stream>>>(conv_b, h);
    int total2 = NGRAPH * OUT_F;
    init_pool_kernel<<<(total2 + 255) / 256, 256, 0, stream>>>(num, denom, gmax);
  }
  // 1) WMMA GEMM: P[N x 2176] = n[N x 64] @ Bmat[64 x 2176]
  {
    int ntiles = NTILE_ROW * NTILE_COL;       // 170000, divisible by 8
    gemm_p_kernel<<<ntiles / 8, 256, 0, stream>>>(n_feat, Bmat, P);
  }
  // 3) per-edge combine + scatter
  edge_kernel<<<NEDGE, 128, 0, stream>>>(e_feat, src, dst, P, h);
  // 4) gate + segment max
  gate_kernel<<<(NNODE + 255) / 256, 256, 0, stream>>>(h, gate_W, gate_b, n2g, gate, gmax);
  gmax_fix_kernel<<<(NGRAPH + 255) / 256, 256, 0, stream>>>(gmax);
  // 5) attention accumulate
  attn_kernel<<<NNODE, 128, 0, stream>>>(h, gate, gmax, n2g, denom, num);
  // 6) pooled -> ELU -> MLP -> out
  final_kernel<<<NGRAPH, 32, 0, stream>>>(num, denom, W1, b1, W2, b2, out);
}